// PhysicsInformedLayer_9122510537269
// MI455X (gfx1250) — compile-verified
//
#include <hip/hip_runtime.h>
#include <hip/hip_bf16.h>

typedef __attribute__((ext_vector_type(2))) float v2f;
typedef __attribute__((ext_vector_type(8))) float v8f;

#define BATCH 32
#define VMIN 0.95f
#define VMAX 1.05f
#define LOSS_BLOCKS 1024

// ---------------------------------------------------------------------------
// Native CDNA5 fp32 scatter atomic: one-shot global_atomic_add_f32 (no return,
// STOREcnt-tracked, device scope -> RMW in L2 atomic units). Avoids the CAS
// retry loop the backend can emit for atomicrmw fadd without metadata.
// ---------------------------------------------------------------------------
__device__ __forceinline__ void atomic_add_f32(float* p, float v)
{
#if defined(__gfx1250__)
    asm volatile("global_atomic_add_f32 %0, %1, off scope:SCOPE_DEV"
                 :: "v"(p), "v"(v) : "memory");
#else
    (void)__hip_atomic_fetch_add(p, v, __ATOMIC_RELAXED, __HIP_MEMORY_SCOPE_AGENT);
#endif
}

// ---------------------------------------------------------------------------
// K1: pass-through copy of v_mag/v_ang to d_out, zero scatter targets + loss.
// ---------------------------------------------------------------------------
__global__ void pf_init_kernel(const float4* __restrict__ vmag,
                               const float4* __restrict__ vang,
                               float4* __restrict__ out_vmag,
                               float4* __restrict__ out_vang,
                               float4* __restrict__ pcalc,
                               float4* __restrict__ qcalc,
                               float* __restrict__ loss,
                               int n4)
{
    int i = blockIdx.x * blockDim.x + threadIdx.x;
    if (i < n4) {
        out_vmag[i] = vmag[i];
        out_vang[i] = vang[i];
        float4 z = make_float4(0.f, 0.f, 0.f, 0.f);
        pcalc[i] = z;
        qcalc[i] = z;
    }
    if (i == 0) *loss = 0.f;
}

// ---------------------------------------------------------------------------
// K2: one thread per edge, inner loop over batches. Gathers are L2-resident
// (v_mag/v_ang = 12.8 MB << 192 MB L2); r/x reads are coalesced per batch
// iteration; scatter via native fp32 L2 atomics. Prefetch next batch slice
// (stride = E*4 bytes = 3.2 MB, beyond any HW streamer reach).
// ---------------------------------------------------------------------------
__global__ void pf_edge_kernel(const float* __restrict__ vmag,
                               const float* __restrict__ vang,
                               const float* __restrict__ rline,
                               const float* __restrict__ xline,
                               const int*   __restrict__ edge_index,
                               float* __restrict__ pcalc,
                               float* __restrict__ qcalc,
                               int N, int E)
{
    int e = blockIdx.x * blockDim.x + threadIdx.x;
    if (e >= E) return;

    const int frm = edge_index[e];
    const int to  = edge_index[E + e];

    for (int b = 0; b < BATCH; ++b) {
        const size_t be  = (size_t)b * E + e;
        const size_t nf  = (size_t)b * N + frm;
        const size_t nt  = (size_t)b * N + to;

        if (b + 1 < BATCH) {
            __builtin_prefetch(&rline[be + E], 0, 1);   // global_prefetch_b8
            __builtin_prefetch(&xline[be + E], 0, 1);
        }

        const float rr = rline[be];
        const float xx = xline[be];
        const float z2 = __fmaf_rn(rr, rr, xx * xx);
        float iz = __builtin_amdgcn_rcpf(z2);           // v_rcp_f32
        iz = iz * __fmaf_rn(-z2, iz, 2.0f);             // 1 Newton step -> ~exact 1/z2
        const float g  =  rr * iz;
        const float bl = -xx * iz;

        const float vi = vmag[nf];
        const float vj = vmag[nt];
        const float th = vang[nf] - vang[nt];

        float st, ct;
        __sincosf(th, &st, &ct);                        // v_sin_f32/v_cos_f32

        const float vij = vi * vj;
        const float pij = vij * __fmaf_rn(g, ct, bl * st);
        const float qij = vij * __fmaf_rn(g, st, -bl * ct);

        atomic_add_f32(&pcalc[nf], pij);
        atomic_add_f32(&qcalc[nf], qij);
    }
}

// ---------------------------------------------------------------------------
// K3: mismatch + loss. Per-thread grid-stride partial, then a wave32
// reduction built on V_WMMA_F32_16X16X4_F32: A holds the 32 lane partials in
// the K=0 (lanes 0-15) / K=2 (lanes 16-31) slots, B = ones(4x16), so
// D[m,n] = acc[m] + acc[m+16]. Summing a lane's 8 D-VGPRs gives the sum of
// rows 0-7 (lanes 0-15) or rows 8-15 (lanes 16-31); lane 0 + lane 16 of each
// wave then carry the full wave sum. EXEC is all-1s at the WMMA (grid-stride
// loop, full blocks, no early return).
// ---------------------------------------------------------------------------
__global__ void pf_loss_kernel(const float* __restrict__ pcalc,
                               const float* __restrict__ qcalc,
                               const float* __restrict__ pbus,
                               const float* __restrict__ qbus,
                               const float* __restrict__ vmag,
                               float* __restrict__ loss,
                               int total)   // B*N
{
    const float wp = 10.0f / (float)total;  // POWER_BALANCE_WEIGHT / (B*N)
    const float wv = 1.0f / (float)BATCH;   // voltage term: sum over N, mean over B

    float acc = 0.f;
    for (int i = blockIdx.x * blockDim.x + threadIdx.x; i < total;
         i += gridDim.x * blockDim.x) {
        const float pm = pcalc[i] - pbus[i];
        const float qm = qcalc[i] - qbus[i];
        const float vm = vmag[i];
        const float lo = fmaxf(VMIN - vm, 0.f);
        const float up = fmaxf(vm - VMAX, 0.f);
        acc += wp * __fmaf_rn(pm, pm, qm * qm)
             + wv * __fmaf_rn(lo, lo, up * up);
    }

    // wave32 reduction via WMMA f32 16x16x4
    v2f a;  a[0] = acc;  a[1] = 0.f;
    v2f bo; bo[0] = 1.f; bo[1] = 1.f;
    v8f c = {};
    v8f d = __builtin_amdgcn_wmma_f32_16x16x4_f32(
        /*neg_a=*/false, a, /*neg_b=*/false, bo,
        /*c_mod=*/(short)0, c, /*reuse_a=*/false, /*reuse_b=*/false);
    const float s = d[0] + d[1] + d[2] + d[3] + d[4] + d[5] + d[6] + d[7];

    __shared__ float red[256];
    red[threadIdx.x] = s;
    __syncthreads();
    if (threadIdx.x == 0) {
        float t = 0.f;
        const int nw = blockDim.x >> 5;
        for (int w = 0; w < nw; ++w)
            t += red[w * 32] + red[w * 32 + 16];
        atomic_add_f32(loss, t);
    }
}

// ---------------------------------------------------------------------------
// Launch
// ---------------------------------------------------------------------------
extern "C" void kernel_launch(void* const* d_in, const int* in_sizes, int n_in,
                              void* d_out, int out_size, void* d_ws, size_t ws_size,
                              hipStream_t stream)
{
    const float* vmag = (const float*)d_in[0];   // (B,N)
    const float* vang = (const float*)d_in[1];   // (B,N)
    const float* rline = (const float*)d_in[2];  // (B,E)
    const float* xline = (const float*)d_in[3];  // (B,E)
    const float* pbus = (const float*)d_in[4];   // (B,N)
    const float* qbus = (const float*)d_in[5];   // (B,N)
    const int* edge_index = (const int*)d_in[6]; // (2,E)

    const int BN = in_sizes[0];        // B*N = 1,600,000
    const int N  = BN / BATCH;
    const int E  = in_sizes[6] / 2;    // 800,000

    float* out      = (float*)d_out;
    float* out_vmag = out;
    float* out_vang = out + (size_t)BN;
    float* loss     = out + 2 * (size_t)BN;

    float* pcalc = (float*)d_ws;           // B*N f32
    float* qcalc = pcalc + (size_t)BN;     // B*N f32

    const int n4 = BN / 4;
    pf_init_kernel<<<(n4 + 255) / 256, 256, 0, stream>>>(
        (const float4*)vmag, (const float4*)vang,
        (float4*)out_vmag, (float4*)out_vang,
        (float4*)pcalc, (float4*)qcalc, loss, n4);

    pf_edge_kernel<<<(E + 255) / 256, 256, 0, stream>>>(
        vmag, vang, rline, xline, edge_index, pcalc, qcalc, N, E);

    pf_loss_kernel<<<LOSS_BLOCKS, 256, 0, stream>>>(
        pcalc, qcalc, pbus, qbus, vmag, loss, BN);
}